// ClassificationBoxProposalProjectionLoss_67860483276817
// MI455X (gfx1250) — compile-verified
//
#include <hip/hip_runtime.h>
#include <stdint.h>

// Problem constants (from the reference)
#define N_IMG   16
#define A_NUM   4096
#define G_NUM   256
#define C_NUM   80
#define BIGV    1.0e8f
#define ALPHA_C 0.25f
#define LOG2E_F 1.4426950408889634f
#define LN2_F   0.6931471805599453f

typedef __attribute__((ext_vector_type(2))) float        v2f;
typedef __attribute__((ext_vector_type(8))) float        v8f;
typedef __attribute__((ext_vector_type(4))) unsigned int u32x4;
typedef __attribute__((ext_vector_type(8))) unsigned int u32x8;

// ---------------------------------------------------------------------------
// TDM: 1-D contiguous tile load Global -> LDS via tensor_load_to_lds.
// Descriptor per CDNA5 ISA (D# group0 + group1, 2-group
// form: tensors up to 2D). nelem must fit in 16 bits (tile_dim0).
// ---------------------------------------------------------------------------
__device__ __forceinline__ void tdm_load_1d(unsigned lds_byte_addr,
                                            const void* gptr,
                                            unsigned nelem,
                                            unsigned log2_data_size) {
  unsigned long long ga = (unsigned long long)(uintptr_t)gptr;
  u32x4 g0;
  g0[0] = 1u;                                                // count = 1
  g0[1] = lds_byte_addr;                                     // lds_addr
  g0[2] = (unsigned)ga;                                      // global_addr lo
  g0[3] = ((unsigned)(ga >> 32) & 0x01FFFFFFu) | (2u << 30); // addr hi | type=2

  u32x8 g1;
  g1[0] = log2_data_size << 16;        // workgroup_mask=0 | data_size
  g1[1] = (nelem & 0xFFFFu) << 16;     // tensor_dim0[15:0]
  g1[2] = (nelem >> 16) | (1u << 16);  // tensor_dim0[31:16] | tensor_dim1=1
  g1[3] = (nelem & 0xFFFFu) << 16;     // tensor_dim1 hi=0 | tile_dim0
  g1[4] = 1u;                          // tile_dim1=1, tile_dim2=0
  g1[5] = nelem;                       // tensor_dim0_stride[31:0]
  g1[6] = 0u;
  g1[7] = 0u;

  asm volatile("tensor_load_to_lds %0, %1" :: "s"(g0), "s"(g1) : "memory");
}

// Sigmoid focal loss term, GAMMA=2, using raw hw transcendentals (exp2/log2
// domain). Safe limits: x<<0 -> t=inf -> rcp=0 -> p=0; x>>0 -> t=0 -> p=1.
__device__ __forceinline__ float focal_term(float x, bool pos) {
  float ax = __builtin_fabsf(x);
  float u  = __builtin_amdgcn_exp2f(-ax * LOG2E_F);   // e^{-|x|} in (0,1]
  float sp = LN2_F * __builtin_amdgcn_logf(1.0f + u); // ln(1 + e^{-|x|})
  float ce = fmaxf(x, 0.0f) - (pos ? x : 0.0f) + sp;  // BCE with logits
  float t  = __builtin_amdgcn_exp2f(-x * LOG2E_F);    // e^{-x}
  float p  = __builtin_amdgcn_rcpf(1.0f + t);         // sigmoid(x)
  float om = pos ? (1.0f - p) : p;                    // 1 - p_t
  float at = pos ? ALPHA_C : (1.0f - ALPHA_C);
  return at * ce * om * om;
}

__global__ __launch_bounds__(256) void
cls_box_proj_loss_kernel(const float* __restrict__ gt_padded,     // [N,G,4]
                         const float* __restrict__ prop_boxes,    // [N,A,4]
                         const float* __restrict__ pred_boxes,    // [N,A,4]
                         const float* __restrict__ class_logits,  // [N,A,C]
                         const unsigned char* __restrict__ gt_masks, // [N,G]
                         const int* __restrict__ gt_classes,      // [N,G]
                         float* __restrict__ out)                 // [2,N,A]
{
  __shared__ float4        s_gt[G_NUM];     // 4 KB, TDM-loaded
  __shared__ int           s_cls[G_NUM];    // 1 KB, TDM-loaded
  __shared__ unsigned char s_msk[G_NUM];    // 256 B, TDM-loaded
  __shared__ int           s_tgt[256];      // per-anchor target class
  __shared__ int           s_any;

  const int tid  = threadIdx.x;
  const int n    = blockIdx.x >> 4;   // image
  const int tile = blockIdx.x & 15;   // 256-anchor tile within image

  if (tid == 0) s_any = 0;

  // --- Phase 0: TDM stage of per-image GT tables into LDS -----------------
  // Every wave issues identical loads (idempotent overlapping writes); each
  // wave waits on its own TENSORcnt, then the workgroup barrier publishes.
  tdm_load_1d((unsigned)(uintptr_t)&s_gt[0],
              gt_padded + (size_t)n * G_NUM * 4, G_NUM * 4, 2u);
  tdm_load_1d((unsigned)(uintptr_t)&s_cls[0],
              gt_classes + (size_t)n * G_NUM, G_NUM, 2u);
  tdm_load_1d((unsigned)(uintptr_t)&s_msk[0],
              gt_masks + (size_t)n * G_NUM, G_NUM, 0u);
  __builtin_amdgcn_s_wait_tensorcnt(0);
  __syncthreads();

  // Pre-mask: poison masked-out GT boxes so the argmin loop needs no mask
  // test. Masked distance ~4e8 never beats a real distance (<=4) when any
  // valid GT exists; in the all-masked case both outputs are zeroed anyway.
  const bool mk = (s_msk[tid] != 0);
  if (mk) atomicOr(&s_any, 1);
  else    s_gt[tid] = make_float4(BIGV, BIGV, BIGV, BIGV);
  __syncthreads();
  const bool nonempty = (s_any != 0);

  // --- Phase 1: per-anchor L1 argmin + projection loss --------------------
  const int    aGlb = tile * 256 + tid;
  const float4 pb   = ((const float4*)prop_boxes)[(size_t)n * A_NUM + aGlb];
  const float4 qb   = ((const float4*)pred_boxes)[(size_t)n * A_NUM + aGlb];

  float best = BIGV;
  int   bi   = 0;
#pragma unroll 4
  for (int g = 0; g < G_NUM; ++g) {
    float4 gb = s_gt[g];
    float d = __builtin_fabsf(pb.x - gb.x) + __builtin_fabsf(pb.y - gb.y) +
              __builtin_fabsf(pb.z - gb.z) + __builtin_fabsf(pb.w - gb.w);
    if (d < best) { best = d; bi = g; }   // strict < keeps first index (argmin)
  }

  {
    float4 gb = s_gt[bi];
    float pd = __builtin_fabsf(qb.x - gb.x) + __builtin_fabsf(qb.y - gb.y) +
               __builtin_fabsf(qb.z - gb.z) + __builtin_fabsf(qb.w - gb.w);
    float proj = s_msk[bi] ? pd : 0.0f;
    out[(size_t)(N_IMG * A_NUM) + (size_t)n * A_NUM + aGlb] = proj;
  }
  s_tgt[tid] = nonempty ? s_cls[bi] : 0;
  __syncthreads();

  // --- Phase 2: focal loss, mean over C via WMMA row-sum ------------------
  // A-matrix (16x4 f32): lanes 0-15 hold K=0,1 for row M=lane; lanes 16-31
  // hold K=2,3 for row M=lane-16. B = ones(4x16) => D accumulates row sums
  // replicated across columns.
  const int lane = tid & 31;
  const int wave = tid >> 5;
  const int half = lane >> 4;   // 0: classes +0,+1   1: classes +2,+3
  const int mrow = lane & 15;   // anchor row within the wave's 16

  v2f bone = {1.0f, 1.0f};

  for (int pass = 0; pass < 2; ++pass) {
    const int abase = pass * 128 + wave * 16;
    const int aL    = abase + mrow;
    const int aG2   = tile * 256 + aL;
    const int tgt   = s_tgt[aL];
    const float* lg = class_logits + ((size_t)n * A_NUM + aG2) * C_NUM;

    v8f acc = {};
#pragma unroll
    for (int kt = 0; kt < C_NUM / 4; ++kt) {
      const int c0 = kt * 4 + half * 2;
      float2 x = *(const float2*)(lg + c0);   // 8B aligned (c0 even)
      v2f amat;
      amat[0] = focal_term(x.x, c0 == tgt);
      amat[1] = focal_term(x.y, (c0 + 1) == tgt);
      acc = __builtin_amdgcn_wmma_f32_16x16x4_f32(
          /*neg_a=*/false, amat, /*neg_b=*/false, bone,
          /*c_mod=*/(short)0, acc, /*reuse_a=*/false, /*reuse_b=*/false);
    }

    // C/D layout: VGPR r of lanes 0-15 = row r; lanes 16-31 = row r+8.
    // Columns identical (B=ones); lanes with mrow<8 each emit one row.
    float r = acc[0];
#pragma unroll
    for (int i = 1; i < 8; ++i) r = (mrow == i) ? acc[i] : r;

    if (mrow < 8) {
      const int row = half * 8 + mrow;
      const int aw  = tile * 256 + abase + row;
      float v = nonempty ? (r * (1.0f / (float)C_NUM)) : 0.0f; // CLS_LAMBDA=1
      out[(size_t)n * A_NUM + aw] = v;
    }
  }
}

extern "C" void kernel_launch(void* const* d_in, const int* in_sizes, int n_in,
                              void* d_out, int out_size, void* d_ws, size_t ws_size,
                              hipStream_t stream) {
  const float*         gt_padded    = (const float*)d_in[0];
  const float*         prop_boxes   = (const float*)d_in[1];
  const float*         pred_boxes   = (const float*)d_in[2];
  const float*         class_logits = (const float*)d_in[3];
  const unsigned char* gt_masks     = (const unsigned char*)d_in[4]; // bool
  const int*           gt_classes   = (const int*)d_in[5];
  float*               out          = (float*)d_out;

  dim3 grid(N_IMG * (A_NUM / 256));  // 256 blocks: 16 images x 16 tiles
  dim3 block(256);                   // 8 wave32s
  hipLaunchKernelGGL(cls_box_proj_loss_kernel, grid, block, 0, stream,
                     gt_padded, prop_boxes, pred_boxes, class_logits,
                     gt_masks, gt_classes, out);
}